// SE3CapsuleLayer_47571057770951
// MI455X (gfx1250) — compile-verified
//
#include <hip/hip_runtime.h>
#include <hip/hip_bf16.h>
#include <math.h>

// ---------------------------------------------------------------------------
// SE(3) capsule routing, MI455X (gfx1250, wave32).
//   K1: exp-map all input poses + transforms to 4x4 row-major matrices in d_ws
//   K2: one WG per (2b x 4m) tile (256 WGs, 512 thr = 16 waves):
//       Phase A: V_WMMA_F32_16X16X4_F32 composes SE(3) pairs (A = 2 P's
//                duplicated, B = 4 T's); 4 n per wave per round so all 32
//                lanes do one se3_log each round; votes kept in LDS.
//       Phase B: 2 waves per (b,m) pair run init-mean + 3 routing iterations
//                + final activation; partials merged via small LDS reduction.
// ---------------------------------------------------------------------------

#define B_  32
#define N_  512
#define M_  64
#define EPS_ 1e-8f

typedef __attribute__((ext_vector_type(2))) float v2f;
typedef __attribute__((ext_vector_type(8))) float v8f;

__device__ __forceinline__ void mat3mul(const float* A, const float* Bm, float* C) {
#pragma unroll
  for (int i = 0; i < 3; ++i)
#pragma unroll
    for (int j = 0; j < 3; ++j)
      C[i * 3 + j] = A[i * 3 + 0] * Bm[0 * 3 + j] +
                     A[i * 3 + 1] * Bm[1 * 3 + j] +
                     A[i * 3 + 2] * Bm[2 * 3 + j];
}

// x[6] = (w, v)  ->  R (3x3 row-major), t (3)
__device__ __forceinline__ void se3_exp6(const float* x, float* R, float* t) {
  float wx = x[0], wy = x[1], wz = x[2];
  float th2 = wx * wx + wy * wy + wz * wz;
  bool  sm  = th2 < 1e-8f;
  float th2s = sm ? 1.f : th2;
  float ths  = sqrtf(th2s);
  float sn = sinf(ths), cs = cosf(ths);
  float A  = sm ? 1.f - th2 * (1.f / 6.f)            : sn / ths;
  float Bc = sm ? 0.5f - th2 * (1.f / 24.f)          : (1.f - cs) / th2s;
  float Cc = sm ? (1.f / 6.f) - th2 * (1.f / 120.f)  : (ths - sn) / (th2s * ths);
  float W[9]  = {0.f, -wz, wy,  wz, 0.f, -wx,  -wy, wx, 0.f};
  float W2[9];
  mat3mul(W, W, W2);
#pragma unroll
  for (int i = 0; i < 9; ++i) R[i] = A * W[i] + Bc * W2[i];
  R[0] += 1.f; R[4] += 1.f; R[8] += 1.f;
  float V[9];
#pragma unroll
  for (int i = 0; i < 9; ++i) V[i] = Bc * W[i] + Cc * W2[i];
  V[0] += 1.f; V[4] += 1.f; V[8] += 1.f;
#pragma unroll
  for (int i = 0; i < 3; ++i)
    t[i] = V[i * 3 + 0] * x[3] + V[i * 3 + 1] * x[4] + V[i * 3 + 2] * x[5];
}

// R (3x3 row-major), t (3)  ->  out[6] = (w, v)
__device__ __forceinline__ void se3_log6(const float* R, const float* t, float* out) {
  float tr = R[0] + R[4] + R[8];
  float cs = fminf(fmaxf((tr - 1.f) * 0.5f, -1.f + 1e-6f), 1.f - 1e-6f);
  float th = acosf(cs);
  bool  sm  = th < 1e-4f;
  float ths = sm ? 1.f : th;
  float v0 = R[7] - R[5], v1 = R[2] - R[6], v2 = R[3] - R[1];
  float fac = sm ? 0.5f + th * th * (1.f / 12.f) : th / (2.f * sinf(ths));
  float w0 = fac * v0, w1 = fac * v1, w2 = fac * v2;
  float coef = sm ? (1.f / 12.f)
                  : (1.f - ths * sinf(ths) / (2.f * (1.f - cosf(ths)))) / (ths * ths);
  float W[9]  = {0.f, -w2, w1,  w2, 0.f, -w0,  -w1, w0, 0.f};
  float W2[9];
  mat3mul(W, W, W2);
  float Vi[9];
#pragma unroll
  for (int i = 0; i < 9; ++i) Vi[i] = -0.5f * W[i] + coef * W2[i];
  Vi[0] += 1.f; Vi[4] += 1.f; Vi[8] += 1.f;
  out[0] = w0; out[1] = w1; out[2] = w2;
#pragma unroll
  for (int i = 0; i < 3; ++i)
    out[3 + i] = Vi[i * 3 + 0] * t[0] + Vi[i * 3 + 1] * t[1] + Vi[i * 3 + 2] * t[2];
}

// ---------------------------------------------------------------------------
// Kernel 1: exponentiate all (B*N) input poses and (M*N) transforms into
// row-major 4x4 matrices [R t; 0 0 0 1] stored in scratch.
// ---------------------------------------------------------------------------
__global__ void se3cap_exp_kernel(const float* __restrict__ poses6,
                                  const float* __restrict__ tij6,
                                  float* __restrict__ expP,
                                  float* __restrict__ expT) {
  int idx = blockIdx.x * blockDim.x + threadIdx.x;
  const int BN = B_ * N_;
  const int MN = M_ * N_;
  if (idx >= BN + MN) return;
  const float* src;
  float* dst;
  if (idx < BN) { src = poses6 + (size_t)idx * 6;        dst = expP + (size_t)idx * 16; }
  else          { src = tij6 + (size_t)(idx - BN) * 6;   dst = expT + (size_t)(idx - BN) * 16; }
  float x[6];
#pragma unroll
  for (int k = 0; k < 6; ++k) x[k] = src[k];
  float R[9], t[3];
  se3_exp6(x, R, t);
  dst[0] = R[0]; dst[1] = R[1]; dst[2]  = R[2]; dst[3]  = t[0];
  dst[4] = R[3]; dst[5] = R[4]; dst[6]  = R[5]; dst[7]  = t[1];
  dst[8] = R[6]; dst[9] = R[7]; dst[10] = R[8]; dst[11] = t[2];
  dst[12] = 0.f; dst[13] = 0.f; dst[14] = 0.f;  dst[15] = 1.f;
}

// ---------------------------------------------------------------------------
// Kernel 2: one workgroup per (2 b's x 4 m's) tile; 512 threads = 16 waves.
// ---------------------------------------------------------------------------
__launch_bounds__(512)
__global__ void se3cap_route_kernel(const float* __restrict__ expP,
                                    const float* __restrict__ expT,
                                    const float* __restrict__ act,
                                    const float* __restrict__ r_alpha,
                                    const float* __restrict__ r_beta,
                                    const float* __restrict__ a_alpha,
                                    const float* __restrict__ a_beta,
                                    float* __restrict__ out) {
  __shared__ float s_votes[8][N_][6];         // 96 KB : votes for 8 pairs
  __shared__ float s_stage[16][4][8][17];     // 34 KB : useful top half of D
  __shared__ float s_act[2][N_];              // 4 KB
  __shared__ float s_red[8][2][7];            // cross-wave partials

  const int b0   = blockIdx.x * 2;   // gridDim.x = 16
  const int m0   = blockIdx.y * 4;   // gridDim.y = 16
  const int tid  = threadIdx.x;
  const int wave = tid >> 5;
  const int lane = tid & 31;

  // cache activations for the 2 b's of this tile
  for (int k = tid; k < 2 * N_; k += 512)
    s_act[k >> 9][k & (N_ - 1)] = act[(size_t)(b0 + (k >> 9)) * N_ + (k & (N_ - 1))];

  // ---------------- Phase A: votes via WMMA ----------------
  // A-frag (16x4 f32): rows = [P(b0) P(b1) P(b0) P(b1)] stacked; lane&15 = row,
  //                    lanes >=16 hold K=2,3.  D rows 8..15 duplicate 0..7.
  // B-frag (4x16 f32): cols = 4 side-by-side T's; lane&15 = col, upper K=2,3.
  const int rl  = lane & 15;
  const int hi  = lane >> 4;
  const int ai  = (rl >> 2) & 1, ar = rl & 3;  // A: which P (dup), which row
  const int bj  = rl >> 2,       bc = rl & 3;  // B: which T, which col
  const int ac0 = hi * 2;                      // A cols {ac0, ac0+1}
  const int br0 = hi * 2;                      // B rows {br0, br0+1}

  for (int rnd = 0; rnd < 8; ++rnd) {
    const int n0 = wave * 32 + rnd * 4;
#pragma unroll
    for (int s = 0; s < 4; ++s) {
      const int n = n0 + s;
      const float* pa = expP + (((size_t)(b0 + ai) * N_ + n) << 4) + ar * 4 + ac0;
      v2f a; a.x = pa[0]; a.y = pa[1];
      const float* pb = expT + (((size_t)(m0 + bj) * N_ + n) << 4) + bc;
      v2f bb; bb.x = pb[br0 * 4]; bb.y = pb[(br0 + 1) * 4];
      v8f c = {};
      v8f d = __builtin_amdgcn_wmma_f32_16x16x4_f32(false, a, false, bb,
                                                    (short)0, c, false, false);
      if (hi == 0) {                 // rows 0..7 hold all useful blocks
#pragma unroll
        for (int r = 0; r < 8; ++r) s_stage[wave][s][r][rl] = d[r];
      }
    }
    __syncthreads();
    // each lane logs one 4x4 block: lane = sel*8 + (i*4 + j), sel = which n
    {
      const int sel = lane >> 3;           // 0..3 -> n0+sel
      const int l   = lane & 7;            // pair index (i*4 + j)
      const int i   = l >> 2, j = l & 3;
      const int n   = n0 + sel;
      float R[9], t[3];
#pragma unroll
      for (int rr = 0; rr < 3; ++rr) {
#pragma unroll
        for (int cc = 0; cc < 3; ++cc)
          R[rr * 3 + cc] = s_stage[wave][sel][4 * i + rr][4 * j + cc];
        t[rr] = s_stage[wave][sel][4 * i + rr][4 * j + 3];
      }
      float v6[6];
      se3_log6(R, t, v6);
#pragma unroll
      for (int c = 0; c < 6; ++c) s_votes[l][n][c] = v6[c];
    }
    __syncthreads();
  }

  // ---------------- Phase B: routing (2 waves per pair) ----------------
  const int p   = wave >> 1;               // pair 0..7
  const int sub = wave & 1;                // half of N handled by this wave
  const int ib  = p >> 2, jm = p & 3;
  const int b   = b0 + ib, m = m0 + jm;
  const int nb  = sub * (N_ / 2);
  const float ral = r_alpha[m], rbe = r_beta[m];
  const float aal = a_alpha[m], abe = a_beta[m];

  float pose[6];
  {
    float s[7] = {0.f, 0.f, 0.f, 0.f, 0.f, 0.f, 0.f};
#pragma unroll
    for (int k = 0; k < 8; ++k) {
      const int n = nb + lane + (k << 5);
      const float a = s_act[ib][n];
#pragma unroll
      for (int c = 0; c < 6; ++c) s[c] += a * s_votes[p][n][c];
      s[6] += a;
    }
#pragma unroll
    for (int off = 16; off > 0; off >>= 1)
#pragma unroll
      for (int c = 0; c < 7; ++c) s[c] += __shfl_xor(s[c], off, 32);
    if (lane == 0)
#pragma unroll
      for (int c = 0; c < 7; ++c) s_red[p][sub][c] = s[c];
    __syncthreads();
    float tot[7];
#pragma unroll
    for (int c = 0; c < 7; ++c) tot[c] = s_red[p][0][c] + s_red[p][1][c];
#pragma unroll
    for (int c = 0; c < 6; ++c) pose[c] = tot[c] / (tot[6] + EPS_);
    __syncthreads();
  }

  for (int it = 0; it < 3; ++it) {
    float Rp[9], tp[3];
    se3_exp6(pose, Rp, tp);
    float Ri[9] = {Rp[0], Rp[3], Rp[6], Rp[1], Rp[4], Rp[7], Rp[2], Rp[5], Rp[8]};
    float ti[3];
#pragma unroll
    for (int i = 0; i < 3; ++i)
      ti[i] = -(Ri[i * 3] * tp[0] + Ri[i * 3 + 1] * tp[1] + Ri[i * 3 + 2] * tp[2]);

    float s[7] = {0.f, 0.f, 0.f, 0.f, 0.f, 0.f, 0.f};
#pragma unroll 2
    for (int k = 0; k < 8; ++k) {
      const int n = nb + lane + (k << 5);
      float v6[6];
#pragma unroll
      for (int c = 0; c < 6; ++c) v6[c] = s_votes[p][n][c];
      float Rv[9], tv[3];
      se3_exp6(v6, Rv, tv);
      float Rc[9];
      mat3mul(Ri, Rv, Rc);
      float tc[3];
#pragma unroll
      for (int i = 0; i < 3; ++i)
        tc[i] = Ri[i * 3] * tv[0] + Ri[i * 3 + 1] * tv[1] + Ri[i * 3 + 2] * tv[2] + ti[i];
      float d6[6];
      se3_log6(Rc, tc, d6);
      float dist = sqrtf(d6[0] * d6[0] + d6[1] * d6[1] + d6[2] * d6[2] +
                         d6[3] * d6[3] + d6[4] * d6[4] + d6[5] * d6[5] + EPS_);
      const float wgt = s_act[ib][n] / (1.f + expf(ral * dist - rbe));
#pragma unroll
      for (int c = 0; c < 6; ++c) s[c] += wgt * v6[c];
      s[6] += wgt;
    }
#pragma unroll
    for (int off = 16; off > 0; off >>= 1)
#pragma unroll
      for (int c = 0; c < 7; ++c) s[c] += __shfl_xor(s[c], off, 32);
    if (lane == 0)
#pragma unroll
      for (int c = 0; c < 7; ++c) s_red[p][sub][c] = s[c];
    __syncthreads();
    float tot[7];
#pragma unroll
    for (int c = 0; c < 7; ++c) tot[c] = s_red[p][0][c] + s_red[p][1][c];
#pragma unroll
    for (int c = 0; c < 6; ++c) pose[c] = tot[c] / (tot[6] + EPS_);
    __syncthreads();
  }

  // final distance pass -> output activation
  float sd = 0.f;
  {
    float Rp[9], tp[3];
    se3_exp6(pose, Rp, tp);
    float Ri[9] = {Rp[0], Rp[3], Rp[6], Rp[1], Rp[4], Rp[7], Rp[2], Rp[5], Rp[8]};
    float ti[3];
#pragma unroll
    for (int i = 0; i < 3; ++i)
      ti[i] = -(Ri[i * 3] * tp[0] + Ri[i * 3 + 1] * tp[1] + Ri[i * 3 + 2] * tp[2]);
#pragma unroll 2
    for (int k = 0; k < 8; ++k) {
      const int n = nb + lane + (k << 5);
      float v6[6];
#pragma unroll
      for (int c = 0; c < 6; ++c) v6[c] = s_votes[p][n][c];
      float Rv[9], tv[3];
      se3_exp6(v6, Rv, tv);
      float Rc[9];
      mat3mul(Ri, Rv, Rc);
      float tc[3];
#pragma unroll
      for (int i = 0; i < 3; ++i)
        tc[i] = Ri[i * 3] * tv[0] + Ri[i * 3 + 1] * tv[1] + Ri[i * 3 + 2] * tv[2] + ti[i];
      float d6[6];
      se3_log6(Rc, tc, d6);
      sd += sqrtf(d6[0] * d6[0] + d6[1] * d6[1] + d6[2] * d6[2] +
                  d6[3] * d6[3] + d6[4] * d6[4] + d6[5] * d6[5] + EPS_);
    }
#pragma unroll
    for (int off = 16; off > 0; off >>= 1) sd += __shfl_xor(sd, off, 32);
    if (lane == 0) s_red[p][sub][0] = sd;
    __syncthreads();
  }

  if (sub == 0 && lane == 0) {
    const float mean_d = (s_red[p][0][0] + s_red[p][1][0]) * (1.f / (float)N_);
    const float oa = 1.f / (1.f + expf(aal * mean_d - abe));
    float* po = out + ((size_t)b * M_ + m) * 6;
#pragma unroll
    for (int c = 0; c < 6; ++c) po[c] = pose[c];
    out[(size_t)B_ * M_ * 6 + (size_t)b * M_ + m] = oa;
  }
}

// ---------------------------------------------------------------------------
extern "C" void kernel_launch(void* const* d_in, const int* in_sizes, int n_in,
                              void* d_out, int out_size, void* d_ws, size_t ws_size,
                              hipStream_t stream) {
  (void)in_sizes; (void)n_in; (void)out_size; (void)ws_size;
  const float* poses6  = (const float*)d_in[0];  // (B,N,6)
  const float* acts    = (const float*)d_in[1];  // (B,N)
  const float* tij6    = (const float*)d_in[2];  // (M,N,6)
  const float* r_alpha = (const float*)d_in[3];  // (1,M,1)
  const float* r_beta  = (const float*)d_in[4];
  const float* a_alpha = (const float*)d_in[5];
  const float* a_beta  = (const float*)d_in[6];
  float* out = (float*)d_out;

  float* expP = (float*)d_ws;                      // B*N*16 floats = 1 MB
  float* expT = expP + (size_t)B_ * N_ * 16;       // M*N*16 floats = 2 MB

  const int total = B_ * N_ + M_ * N_;
  se3cap_exp_kernel<<<(total + 255) / 256, 256, 0, stream>>>(poses6, tij6, expP, expT);

  dim3 grid(B_ / 2, M_ / 4);   // 16 x 16 tiles of (2 b's x 4 m's)
  se3cap_route_kernel<<<grid, 512, 0, stream>>>(expP, expT, acts,
                                                r_alpha, r_beta, a_alpha, a_beta, out);
}